// MatchNet_773094113749
// MI455X (gfx1250) — compile-verified
//
#include <hip/hip_runtime.h>
#include <math.h>

#define WAVES 8
#define PDHG_ITERS 150
#define CSTR 10.0f

typedef __attribute__((ext_vector_type(2))) float v2f;
typedef __attribute__((ext_vector_type(8))) float v8f;

__device__ __forceinline__ v8f vzero() {
  v8f z = {0.f, 0.f, 0.f, 0.f, 0.f, 0.f, 0.f, 0.f};
  return z;
}

// D = A(16x4) * B(4x16) + C, full fp32 via CDNA5 WMMA.
__device__ __forceinline__ v8f wmma4(v2f a, v2f b, v8f c) {
  return __builtin_amdgcn_wmma_f32_16x16x4_f32(false, a, false, b, (short)0, c,
                                               false, false);
}

// Branch-free tanh: 1 - 2/(e^{2x}+1). v_exp_f32 + v_rcp_f32, no EXEC-mask
// branches (keeps EXEC all-ones around WMMA issue). Saturates to +-1.
__device__ __forceinline__ float fast_tanh(float x) {
  float e = __expf(2.0f * x);
  return 1.0f - 2.0f * __builtin_amdgcn_rcpf(e + 1.0f);
}

// Intra-wave LDS ordering: producers (ds stores from some lanes) must be
// visible to consumers (ds loads in other lanes of the same wave).
__device__ __forceinline__ void wave_lds_fence() {
  __builtin_amdgcn_wave_barrier();
  asm volatile("s_wait_dscnt 0x0" ::: "memory");
  __builtin_amdgcn_wave_barrier();
}

// A fragment (16x4 f32): lane m=lane%16 holds A[m][k0+2*hi], A[m][k0+2*hi+1]
__device__ __forceinline__ v2f ldA(const float* b, int ld, int m, int k0, int hi) {
  const float* p = b + m * ld + k0 + 2 * hi;
  v2f r;
  r.x = p[0];
  r.y = p[1];
  return r;
}
// B fragment (4x16 f32): lane n=lane%16 holds B[k0+2*hi][n], B[k0+2*hi+1][n]
__device__ __forceinline__ v2f ldB(const float* b, int ld, int k0, int nn, int hi) {
  int k = k0 + 2 * hi;
  v2f r;
  r.x = b[k * ld + nn];
  r.y = b[(k + 1) * ld + nn];
  return r;
}

__global__ __launch_bounds__(32 * WAVES) void matchnet_kernel(
    const float* __restrict__ X,
    const float* __restrict__ W1, const float* __restrict__ B1,
    const float* __restrict__ W2, const float* __restrict__ B2,
    const float* __restrict__ W3, const float* __restrict__ B3,
    const float* __restrict__ W4, const float* __restrict__ B4,
    const float* __restrict__ S,
    float* __restrict__ out, int ntiles, int batch) {
  // Block-shared, zero-padded weight/operator tiles.
  __shared__ float s_w1[8 * 32];    // [K=8 pad][N=32 pad] (6x20 real)
  __shared__ float s_w2[20 * 32];   // 20x20 real
  __shared__ float s_w3[20 * 32];
  __shared__ float s_w4[20 * 16];   // 20x8 real
  __shared__ float s_b1[32], s_b2[32], s_b3[32], s_b4[16];
  __shared__ float s_Ap[16 * 16];   // A=[S;-I] padded 14x8 -> 16x16
  __shared__ float s_ATp[8 * 16];   // A^T padded 8x14 -> 8x16
  // Per-wave tiles.
  __shared__ float s_z[WAVES][16 * 8];    // Z tile (cols 6,7 zero)
  __shared__ float s_hA[WAVES][16 * 32];  // activation ping
  __shared__ float s_hB[WAVES][16 * 32];  // activation pong
  __shared__ float s_xb[WAVES][16 * 8];   // xbar tile
  __shared__ float s_y[WAVES][16 * 16];   // dual y tile (cols 14,15 zero)

  const int tid = threadIdx.x;
  const int nth = blockDim.x;

  for (int i = tid; i < 8 * 32; i += nth) {
    int r = i >> 5, c = i & 31;
    s_w1[i] = (r < 6 && c < 20) ? W1[r * 20 + c] : 0.f;
  }
  for (int i = tid; i < 20 * 32; i += nth) {
    int r = i >> 5, c = i & 31;
    s_w2[i] = (c < 20) ? W2[r * 20 + c] : 0.f;
    s_w3[i] = (c < 20) ? W3[r * 20 + c] : 0.f;
  }
  for (int i = tid; i < 20 * 16; i += nth) {
    int r = i >> 4, c = i & 15;
    s_w4[i] = (c < 8) ? W4[r * 8 + c] : 0.f;
  }
  for (int i = tid; i < 32; i += nth) {
    s_b1[i] = (i < 20) ? B1[i] : 0.f;
    s_b2[i] = (i < 20) ? B2[i] : 0.f;
    s_b3[i] = (i < 20) ? B3[i] : 0.f;
  }
  for (int i = tid; i < 16; i += nth) s_b4[i] = (i < 8) ? B4[i] : 0.f;
  for (int i = tid; i < 256; i += nth) {
    int r = i >> 4, c = i & 15;
    float v = 0.f;
    if (r < 6 && c < 8) v = S[r * 8 + c];
    else if (r >= 6 && r < 14 && c == (r - 6)) v = -1.f;
    s_Ap[i] = v;
  }
  for (int i = tid; i < 128; i += nth) {
    int k = i >> 4, c = i & 15;
    float v = 0.f;
    if (c < 6) v = S[c * 8 + k];
    else if (c < 14 && k == (c - 6)) v = -1.f;
    s_ATp[i] = v;
  }
  __syncthreads();

  // tau = sigma = 1/||A||_F, A = [S; -I8]
  float ss = 8.0f;
#pragma unroll
  for (int i = 0; i < 48; ++i) {
    float v = S[i];
    ss = fmaf(v, v, ss);
  }
  const float tau = 1.0f / sqrtf(ss);
  const float sigma = tau;

  const int wave = tid >> 5;
  const int lane = tid & 31;
  const int hi = lane >> 4;   // half-wave select
  const int nn = lane & 15;   // A-frag row / B,C,D-frag column
  const int tile = blockIdx.x * WAVES + wave;
  if (tile >= ntiles) return;
  const int row0 = tile * 16;

  float* zt = s_z[wave];
  float* hA = s_hA[wave];
  float* hB = s_hB[wave];
  float* xbt = s_xb[wave];
  float* yt = s_y[wave];

  // Stage Z tile [16 x 8] (cols 6,7 zero).
  for (int i = lane; i < 128; i += 32) {
    int r = i >> 3, c = i & 7;
    int rg = row0 + r;
    zt[i] = (c < 6 && rg < batch) ? X[rg * 6 + c] : 0.f;
  }
  wave_lds_fence();

  // ---------------- MLP (all GEMMs on the matrix pipe) ----------------
  v8f h0 = vzero(), h1 = vzero();
#pragma unroll
  for (int k0 = 0; k0 < 8; k0 += 4) {
    v2f a = ldA(zt, 8, nn, k0, hi);
    h0 = wmma4(a, ldB(s_w1, 32, k0, nn, hi), h0);
    h1 = wmma4(a, ldB(s_w1 + 16, 32, k0, nn, hi), h1);
  }
#pragma unroll
  for (int v = 0; v < 8; ++v) {
    int r = v + 8 * hi;
    hA[r * 32 + nn] = fast_tanh(h0[v] + s_b1[nn]);
    hA[r * 32 + 16 + nn] = fast_tanh(h1[v] + s_b1[16 + nn]);  // pads -> tanh(0)=0
  }
  wave_lds_fence();

  h0 = vzero();
  h1 = vzero();
#pragma unroll
  for (int k0 = 0; k0 < 20; k0 += 4) {
    v2f a = ldA(hA, 32, nn, k0, hi);
    h0 = wmma4(a, ldB(s_w2, 32, k0, nn, hi), h0);
    h1 = wmma4(a, ldB(s_w2 + 16, 32, k0, nn, hi), h1);
  }
#pragma unroll
  for (int v = 0; v < 8; ++v) {
    int r = v + 8 * hi;
    hB[r * 32 + nn] = fast_tanh(h0[v] + s_b2[nn]);
    hB[r * 32 + 16 + nn] = fast_tanh(h1[v] + s_b2[16 + nn]);
  }
  wave_lds_fence();

  h0 = vzero();
  h1 = vzero();
#pragma unroll
  for (int k0 = 0; k0 < 20; k0 += 4) {
    v2f a = ldA(hB, 32, nn, k0, hi);
    h0 = wmma4(a, ldB(s_w3, 32, k0, nn, hi), h0);
    h1 = wmma4(a, ldB(s_w3 + 16, 32, k0, nn, hi), h1);
  }
#pragma unroll
  for (int v = 0; v < 8; ++v) {
    int r = v + 8 * hi;
    hA[r * 32 + nn] = fast_tanh(h0[v] + s_b3[nn]);
    hA[r * 32 + 16 + nn] = fast_tanh(h1[v] + s_b3[16 + nn]);
  }
  wave_lds_fence();

  v8f zacc = vzero();
#pragma unroll
  for (int k0 = 0; k0 < 20; k0 += 4)
    zacc = wmma4(ldA(hA, 32, nn, k0, hi), ldB(s_w4, 16, k0, nn, hi), zacc);

  v8f zreg;
#pragma unroll
  for (int v = 0; v < 8; ++v)
    zreg[v] = (nn < 8) ? (zacc[v] + s_b4[nn]) : 0.f;

  // bfull tile in D layout: cols 0..5 = Z row, rest 0.
  v8f bf;
#pragma unroll
  for (int v = 0; v < 8; ++v)
    bf[v] = (nn < 6) ? zt[(v + 8 * hi) * 8 + nn] : 0.f;

  // ---------------- PDHG ----------------
  v8f x, y = vzero();
#pragma unroll
  for (int v = 0; v < 8; ++v) x[v] = fmaxf(zreg[v], 0.f);
  if (nn < 8) {
#pragma unroll
    for (int v = 0; v < 8; ++v) xbt[(v + 8 * hi) * 8 + nn] = x[v];
  }
  wave_lds_fence();

  // Loop-invariant B fragments for A^T (8x16) and A (16x16).
  const v2f bAT0 = ldB(s_ATp, 16, 0, nn, hi);
  const v2f bAT1 = ldB(s_ATp, 16, 4, nn, hi);
  const v2f bA0 = ldB(s_Ap, 16, 0, nn, hi);
  const v2f bA1 = ldB(s_Ap, 16, 4, nn, hi);
  const v2f bA2 = ldB(s_Ap, 16, 8, nn, hi);
  const v2f bA3 = ldB(s_Ap, 16, 12, nn, hi);

  const float tc = tau * CSTR;

#pragma unroll 1
  for (int it = 0; it < PDHG_ITERS; ++it) {
    // u = xbar @ A^T   (16x14 result, cols 14,15 stay 0)
    v8f u = vzero();
    u = wmma4(ldA(xbt, 8, nn, 0, hi), bAT0, u);
    u = wmma4(ldA(xbt, 8, nn, 4, hi), bAT1, u);
#pragma unroll
    for (int v = 0; v < 8; ++v) {
      y[v] = fmaxf(0.f, y[v] + sigma * (u[v] - bf[v]));
      yt[(v + 8 * hi) * 16 + nn] = y[v];
    }
    wave_lds_fence();

    // w = y @ A   (16x8 result)
    v8f w = vzero();
    w = wmma4(ldA(yt, 16, nn, 0, hi), bA0, w);
    w = wmma4(ldA(yt, 16, nn, 4, hi), bA1, w);
    w = wmma4(ldA(yt, 16, nn, 8, hi), bA2, w);
    w = wmma4(ldA(yt, 16, nn, 12, hi), bA3, w);

    // prox_g + over-relaxation; row L2-norm over lanes 0..7 (and 16..23).
#pragma unroll
    for (int v = 0; v < 8; ++v) {
      float d = (x[v] - tau * w[v] + tau) - zreg[v];
      float s2 = d * d;
      s2 += __shfl_xor(s2, 1, 32);
      s2 += __shfl_xor(s2, 2, 32);
      s2 += __shfl_xor(s2, 4, 32);
      float nrm = __builtin_amdgcn_sqrtf(s2);
      float scl =
          fmaxf(0.f, 1.f - tc * __builtin_amdgcn_rcpf(fmaxf(nrm, 1e-12f)));
      float xn = (nn < 8) ? (zreg[v] + scl * d) : 0.f;
      float xb = 2.f * xn - x[v];
      x[v] = xn;
      if (nn < 8) xbt[(v + 8 * hi) * 8 + nn] = xb;
    }
    wave_lds_fence();
  }

  if (nn < 8) {
#pragma unroll
    for (int v = 0; v < 8; ++v) {
      int rg = row0 + v + 8 * hi;
      if (rg < batch) out[rg * 8 + nn] = x[v];
    }
  }
}

extern "C" void kernel_launch(void* const* d_in, const int* in_sizes, int n_in,
                              void* d_out, int out_size, void* d_ws,
                              size_t ws_size, hipStream_t stream) {
  const float* X = (const float*)d_in[0];
  const float* W1 = (const float*)d_in[1];
  const float* B1 = (const float*)d_in[2];
  const float* W2 = (const float*)d_in[3];
  const float* B2 = (const float*)d_in[4];
  const float* W3 = (const float*)d_in[5];
  const float* B3 = (const float*)d_in[6];
  const float* W4 = (const float*)d_in[7];
  const float* B4 = (const float*)d_in[8];
  const float* S = (const float*)d_in[9];
  float* out = (float*)d_out;

  int batch = in_sizes[0] / 6;
  int ntiles = (batch + 15) / 16;
  int nblocks = (ntiles + WAVES - 1) / WAVES;

  matchnet_kernel<<<nblocks, 32 * WAVES, 0, stream>>>(
      X, W1, B1, W2, B2, W3, B3, W4, B4, S, out, ntiles, batch);
}